// GCNPlusConv_27419071218309
// MI455X (gfx1250) — compile-verified
//
#include <hip/hip_runtime.h>

#define D 128
#define KHOPS 10

typedef float v2f __attribute__((ext_vector_type(2)));
typedef float v8f __attribute__((ext_vector_type(8)));

// ---------------- coefficients: coeffs[k] = e^{-t} * t^k / k! ----------------
__global__ void coeffs_kernel(const float* __restrict__ t_ptr, float* __restrict__ coeffs) {
    if (threadIdx.x == 0 && blockIdx.x == 0) {
        float t = t_ptr[0];
        float c = expf(-t);
        coeffs[0] = c;
        for (int k = 1; k <= KHOPS; ++k) { c = c * t / (float)k; coeffs[k] = c; }
    }
}

// ---------------- degree (self-loop seeds deg=1) ----------------
__global__ void deg_init_kernel(float* __restrict__ deg, int N) {
    int i = blockIdx.x * blockDim.x + threadIdx.x;
    if (i < N) deg[i] = 1.0f;
}

__global__ void deg_count_kernel(const long long* __restrict__ ei, float* __restrict__ deg, int E) {
    int e = blockIdx.x * blockDim.x + threadIdx.x;
    if (e < E) {
        int d = (int)ei[(long long)E + e];   // edge_index[1][e]
        atomicAdd(&deg[d], 1.0f);
    }
}

__global__ void dinv_kernel(float* __restrict__ deg, int N) {
    int i = blockIdx.x * blockDim.x + threadIdx.x;
    if (i < N) deg[i] = rsqrtf(deg[i]);      // deg >= 1 always (self-loop)
}

// ---------------- h = x ; acc = coeffs[0] * x ----------------
__global__ void init_kernel(const float* __restrict__ x, float* __restrict__ h,
                            float* __restrict__ acc, const float* __restrict__ coeffs,
                            int total4) {
    int i = blockIdx.x * blockDim.x + threadIdx.x;
    if (i < total4) {
        float4 v = ((const float4*)x)[i];
        ((float4*)h)[i] = v;
        float c = coeffs[0];
        float4 o; o.x = c * v.x; o.y = c * v.y; o.z = c * v.z; o.w = c * v.w;
        ((float4*)acc)[i] = o;
    }
}

// ---------------- h2 = dinv[n]^2 * h  (self-loop term, also zero-init replacement) ----------------
__global__ void selfloop_kernel(const float* __restrict__ h, const float* __restrict__ dinv,
                                float* __restrict__ h2, int total4) {
    int i = blockIdx.x * blockDim.x + threadIdx.x;
    if (i < total4) {
        int n = i >> 5;                       // 32 float4 per 128-float row
        float w = dinv[n]; w = w * w;
        float4 v = ((const float4*)h)[i];
        float4 o; o.x = w * v.x; o.y = w * v.y; o.z = w * v.z; o.w = w * v.w;
        ((float4*)h2)[i] = o;
    }
}

// ---------------- one wave per edge: h2[dst] += dinv[s]*dinv[d] * h[src] ----------------
__global__ void __launch_bounds__(256)
prop_kernel(const long long* __restrict__ ei, const float* __restrict__ dinv,
            const float* __restrict__ h, float* __restrict__ h2, int E) {
    int gid  = blockIdx.x * blockDim.x + threadIdx.x;
    int e    = gid >> 5;
    int lane = gid & 31;
    if (e >= E) return;                       // uniform per wave
    int s = (int)ei[e];
    int d = (int)ei[(long long)E + e];
    float w = dinv[s] * dinv[d];
    float4 v = ((const float4*)(h + (size_t)s * D))[lane];   // global_load_b128, 512B/row coalesced
    float* o = h2 + (size_t)d * D + lane * 4;
    atomicAdd(o + 0, w * v.x);
    atomicAdd(o + 1, w * v.y);
    atomicAdd(o + 2, w * v.z);
    atomicAdd(o + 3, w * v.w);
}

// ---------------- acc += coeffs[k] * h2 ----------------
__global__ void axpy_kernel(float* __restrict__ acc, const float* __restrict__ h2,
                            const float* __restrict__ coeffs, int k, int total4) {
    int i = blockIdx.x * blockDim.x + threadIdx.x;
    if (i < total4) {
        float c = coeffs[k];
        float4 a = ((float4*)acc)[i];
        float4 v = ((const float4*)h2)[i];
        a.x += c * v.x; a.y += c * v.y; a.z += c * v.z; a.w += c * v.w;
        ((float4*)acc)[i] = a;
    }
}

// ---------------- out[i,:] = acc[i,:] @ W^T + b  via V_WMMA_F32_16X16X4_F32 ----------------
// Block: 256 threads = 8 waves. Each block -> 16 output rows; wave w -> cols [16w, 16w+16).
// In-place safe: A tile fully staged in LDS before stores.
__global__ void __launch_bounds__(256)
matmul_wmma_kernel(const float* __restrict__ acc, const float* __restrict__ Wm,
                   const float* __restrict__ bias, float* __restrict__ out, int N) {
    __shared__ float As[16][132];             // pad 128->132: kills bank conflicts on column reads
    int rowbase = blockIdx.x * 16;
    int tid = threadIdx.x;

    // stage 16x128 fp32 A tile (512 float4, 2 per thread)
    for (int it = 0; it < 2; ++it) {
        int idx = tid + it * 256;             // 0..511
        int r   = idx >> 5;                   // tile row 0..15
        int c4  = idx & 31;                   // float4 column
        int row = rowbase + r; if (row > N - 1) row = N - 1;   // clamp, no branch
        float4 v = ((const float4*)(acc + (size_t)row * D))[c4];
        As[r][c4 * 4 + 0] = v.x;
        As[r][c4 * 4 + 1] = v.y;
        As[r][c4 * 4 + 2] = v.z;
        As[r][c4 * 4 + 3] = v.w;
    }
    __syncthreads();

    int wave = tid >> 5;
    int lane = tid & 31;
    int half = lane >> 4;                     // 0: lanes 0-15, 1: lanes 16-31
    int ln   = lane & 15;
    int colbase = wave * 16;

    v8f c = {};
    #pragma unroll
    for (int k0 = 0; k0 < D; k0 += 4) {
        // A 16x4 layout: lanes 0-15 hold K=k0,k0+1; lanes 16-31 hold K=k0+2,k0+3 (M = ln)
        v2f a;
        a.x = As[ln][k0 + 2 * half];
        a.y = As[ln][k0 + 2 * half + 1];
        // B 4x16 layout: VGPR0 = K rows {0,2}, VGPR1 = K rows {1,3}; N = ln.
        // B[kk][n] = W[colbase+n][k0+kk]
        const float* wr = Wm + (size_t)(colbase + ln) * D + k0 + 2 * half;
        v2f b;
        b.x = wr[0];
        b.y = wr[1];
        c = __builtin_amdgcn_wmma_f32_16x16x4_f32(false, a, false, b, (short)0, c, false, false);
    }

    // C/D layout: VGPR r -> M = r + 8*half, N = ln
    #pragma unroll
    for (int r = 0; r < 8; ++r) {
        int row = rowbase + r + 8 * half;
        int col = colbase + ln;
        if (row < N) out[(size_t)row * D + col] = c[r] + bias[col];
    }
}

extern "C" void kernel_launch(void* const* d_in, const int* in_sizes, int n_in,
                              void* d_out, int out_size, void* d_ws, size_t ws_size,
                              hipStream_t stream) {
    const float*     x    = (const float*)d_in[0];
    const long long* ei   = (const long long*)d_in[1];   // int64 edge_index [2,E]
    const float*     t    = (const float*)d_in[2];
    const float*     Wm   = (const float*)d_in[3];
    const float*     bias = (const float*)d_in[4];
    float*           out  = (float*)d_out;

    int N = in_sizes[0] / D;
    int E = in_sizes[1] / 2;
    size_t nd = (size_t)N * D;
    int total4 = (int)(nd / 4);

    // workspace layout (floats): coeffs[16] | deg[N] | h[N*D] | h2[N*D]
    float* ws     = (float*)d_ws;
    float* coeffs = ws;
    float* deg    = ws + 16;
    float* h      = deg + N;
    float* h2     = h + nd;

    coeffs_kernel<<<1, 1, 0, stream>>>(t, coeffs);
    deg_init_kernel<<<(N + 255) / 256, 256, 0, stream>>>(deg, N);
    deg_count_kernel<<<(E + 255) / 256, 256, 0, stream>>>(ei, deg, E);
    dinv_kernel<<<(N + 255) / 256, 256, 0, stream>>>(deg, N);
    init_kernel<<<(total4 + 255) / 256, 256, 0, stream>>>(x, h, out, coeffs, total4);

    for (int k = 1; k <= KHOPS; ++k) {
        selfloop_kernel<<<(total4 + 255) / 256, 256, 0, stream>>>(h, deg, h2, total4);
        long long threads = (long long)E * 32;
        int blocks = (int)((threads + 255) / 256);
        prop_kernel<<<blocks, 256, 0, stream>>>(ei, deg, h, h2, E);
        axpy_kernel<<<(total4 + 255) / 256, 256, 0, stream>>>(out, h2, coeffs, k, total4);
        float* tmp = h; h = h2; h2 = tmp;
    }

    matmul_wmma_kernel<<<(N + 15) / 16, 256, 0, stream>>>(out, Wm, bias, out, N);
}